// Parallel_DFS_SE_36309653520964
// MI455X (gfx1250) — compile-verified
//
#include <hip/hip_runtime.h>
#include <hip/hip_bf16.h>

#define NS 32
#define CH 256
#define HD 128
#define NP 3136
#define TS 21

typedef float v2f __attribute__((ext_vector_type(2)));
typedef float v4f __attribute__((ext_vector_type(4)));
typedef float v8f __attribute__((ext_vector_type(8)));

// ---------------- LDS layout for the scan kernel (float offsets) ------------
#define OFF_NODEW 0                 // 128 rows x 258 (padded)   = 33024
#define OFF_POL1W 33024             // 128 rows x 130 (padded)   = 16640
#define OFF_XS    49664             // 32 x 258                  = 8256
#define OFF_H     57920             // 32 x 130                  = 4160
#define OFF_P1    62080             // 32 x 130                  = 4160
#define OFF_PATH  66240             // 32 x 128                  = 4096
#define OFF_NODEB 70336             // 128
#define OFF_POL1B 70464             // 128
#define OFF_POL2W 70592             // 9 x 128                   = 1152
#define OFF_POL2B 71744             // 16
#define OFF_LOGIT 71760             // 32 x 12                   = 384
#define OFF_WPREV 72144             // 32
#define OFF_CURR  72176             // 32 (int view)
#define OFF_VIS   72208             // 32 x 98 (uint view)       = 3136
#define SMEM_FLOATS 75344
#define SMEM_BYTES  (SMEM_FLOATS * 4)

// ---------------------------------------------------------------------------
// Kernel A: per-(n,c) spatial mean of x.  One block per row, vec4 streaming.
// Default (RT) loads deliberately fill the 192MB L2: x (103MB) stays resident
// so k_scale's second pass reads from L2 instead of HBM.
// ---------------------------------------------------------------------------
__global__ void k_means(const float* __restrict__ x, float* __restrict__ s_g) {
    __shared__ float red[256];
    const int row = blockIdx.x;                       // n*256 + c
    const v4f* xr = (const v4f*)(x + (size_t)row * NP);
    float acc = 0.f;
    for (int i = threadIdx.x; i < NP / 4; i += 256) {
        v4f v = xr[i];
        acc += v.x + v.y + v.z + v.w;
    }
    red[threadIdx.x] = acc;
    __syncthreads();
    for (int off = 128; off > 0; off >>= 1) {
        if (threadIdx.x < off) red[threadIdx.x] += red[threadIdx.x + off];
        __syncthreads();
    }
    if (threadIdx.x == 0) s_g[row] = red[0] * (1.f / (float)NP);
}

// ---------------------------------------------------------------------------
// Kernel B: both SE attentions per sample. One block per n.
// ---------------------------------------------------------------------------
__device__ __forceinline__ float sigmoidf(float z) { return 1.f / (1.f + __expf(-z)); }

__global__ void k_se(const float* __restrict__ s_g,
                     const float* __restrict__ se1_w, const float* __restrict__ se1_b,
                     const float* __restrict__ se2_w, const float* __restrict__ se2_b,
                     const float* __restrict__ bse1_w, const float* __restrict__ bse1_b,
                     const float* __restrict__ bse2_w, const float* __restrict__ bse2_b,
                     float* __restrict__ attse_g, float* __restrict__ attse2_g) {
    __shared__ float sv[CH];
    __shared__ float h1[16];
    const int n = blockIdx.x, tid = threadIdx.x;
    sv[tid] = s_g[n * CH + tid];
    __syncthreads();

    // --- first SE ---
    if (tid < 16) {
        float a = se1_b[tid];
        const float* wr = se1_w + tid * CH;
        for (int c = 0; c < CH; ++c) a += sv[c] * wr[c];
        h1[tid] = fmaxf(a, 0.f);
    }
    __syncthreads();
    {
        float a = se2_b[tid];
        const float* wr = se2_w + tid * 16;
        for (int j = 0; j < 16; ++j) a += h1[j] * wr[j];
        attse_g[n * CH + tid] = sigmoidf(a);
    }
    __syncthreads();
    // --- second SE (bse) ---
    if (tid < 16) {
        float a = bse1_b[tid];
        const float* wr = bse1_w + tid * CH;
        for (int c = 0; c < CH; ++c) a += sv[c] * wr[c];
        h1[tid] = fmaxf(a, 0.f);
    }
    __syncthreads();
    {
        float a = bse2_b[tid];
        const float* wr = bse2_w + tid * 16;
        for (int j = 0; j < 16; ++j) a += h1[j] * wr[j];
        attse2_g[n * CH + tid] = sigmoidf(a);
    }
}

// ---------------------------------------------------------------------------
// 32xK @ KxN(=128) GEMM on f32 WMMA 16x16x4 tiles. 256 threads = 8 waves;
// wave w owns output columns [w*16, w*16+16); two M tiles (rows 0-15, 16-31).
// B is the weight matrix stored [out][in] row-major (out = output column).
// Must be called by all 256 threads with uniform control flow (EXEC all 1s).
// ---------------------------------------------------------------------------
__device__ __forceinline__ void gemm32(const float* __restrict__ A, int as,
                                       const float* __restrict__ B, int bs,
                                       int K,
                                       float* __restrict__ D, int ds,
                                       const float* __restrict__ bias,
                                       int relu, int tid) {
    const int w    = tid >> 5;
    const int lane = tid & 31;
    const int ll   = lane & 15;
    const int hi2  = (lane >> 4) << 1;   // K sub-offset {0,2}
    const int hr   = (lane >> 4) << 3;   // D row sub-offset {0,8}

    v8f c0 = {0.f, 0.f, 0.f, 0.f, 0.f, 0.f, 0.f, 0.f};
    v8f c1 = c0;

    const float* Bp = B + (w * 16 + ll) * bs + hi2;
    const float* A0 = A + ll * as + hi2;
    const float* A1 = A + (16 + ll) * as + hi2;

    for (int k = 0; k < K; k += 4) {
        v2f b  = *(const v2f*)(Bp + k);
        v2f a0 = *(const v2f*)(A0 + k);
        v2f a1 = *(const v2f*)(A1 + k);
        c0 = __builtin_amdgcn_wmma_f32_16x16x4_f32(false, a0, false, b, (short)0, c0, false, false);
        c1 = __builtin_amdgcn_wmma_f32_16x16x4_f32(false, a1, false, b, (short)0, c1, false, false);
    }
    const float bb = bias ? bias[w * 16 + ll] : 0.f;
    for (int r = 0; r < 8; ++r) {
        float v0 = c0[r] + bb;
        float v1 = c1[r] + bb;
        if (relu) { v0 = fmaxf(v0, 0.f); v1 = fmaxf(v1, 0.f); }
        D[(r + hr) * ds + w * 16 + ll]      = v0;
        D[(16 + r + hr) * ds + w * 16 + ll] = v1;
    }
}

// ---------------------------------------------------------------------------
// Kernel C: full 21-step DFS scan for all 32 samples in one workgroup.
// Node features are computed on demand at each walker's position, so the
// reference's (N,P,128) einsum never materializes.
// ---------------------------------------------------------------------------
__global__ void k_scan(const float* __restrict__ x,
                       const float* __restrict__ gu,
                       const float* __restrict__ node_w, const float* __restrict__ node_b,
                       const float* __restrict__ pol1_w, const float* __restrict__ pol1_b,
                       const float* __restrict__ pol2_w, const float* __restrict__ pol2_b,
                       const float* __restrict__ dfs_w,  const float* __restrict__ dfs_b,
                       const float* __restrict__ fusion_w,
                       const float* __restrict__ attse_g,
                       const float* __restrict__ attse2_g,
                       float* __restrict__ att_total_g) {
    extern __shared__ float smem[];
    const int tid = threadIdx.x;
    int*      currL = (int*)&smem[OFF_CURR];
    unsigned* visL  = (unsigned*)&smem[OFF_VIS];

    // ---- init state + stage weights in LDS (reused for all 22 GEMM rounds)
    for (int i = tid; i < NS * 98; i += 256) visL[i] = 0u;
    for (int i = tid; i < NS * HD; i += 256) smem[OFF_PATH + i] = 0.f;
    if (tid < NS) { currL[tid] = 0; smem[OFF_WPREV + tid] = 0.f; }
    for (int i = tid; i < HD * CH; i += 256)
        smem[OFF_NODEW + (i >> 8) * 258 + (i & 255)] = node_w[i];
    for (int i = tid; i < HD * HD; i += 256)
        smem[OFF_POL1W + (i >> 7) * 130 + (i & 127)] = pol1_w[i];
    if (tid < HD) { smem[OFF_NODEB + tid] = node_b[tid]; smem[OFF_POL1B + tid] = pol1_b[tid]; }
    for (int i = tid; i < 9 * HD; i += 256) smem[OFF_POL2W + i] = pol2_w[i];
    if (tid < 9) smem[OFF_POL2B + tid] = pol2_b[tid];
    __syncthreads();

    const int dy[8] = {-1, -1, -1, 0, 0, 1, 1, 1};
    const int dx[8] = {-1, 0, 1, -1, 1, -1, 0, 1};

    for (int t = 0; t <= TS; ++t) {
        // ---- gather x columns at current positions, scaled by att_se
        for (int i = tid; i < NS * CH; i += 256) {
            const int n = i >> 8, c = i & 255;
            smem[OFF_XS + n * 258 + c] =
                x[(size_t)(n * CH + c) * NP + currL[n]] * attse_g[i];
        }
        __syncthreads();

        // ---- node features H = Xs(32x256) @ node_w^T + node_b  (WMMA f32)
        gemm32(&smem[OFF_XS], 258, &smem[OFF_NODEW], 258, CH,
               &smem[OFF_H], 130, &smem[OFF_NODEB], 0, tid);
        __syncthreads();

        // ---- path_sum += (1-stop_{t-1}) * node_flat[nxt_{t-1}]  (wprev==0 at t=0)
        for (int i = tid; i < NS * HD; i += 256) {
            const int n = i >> 7, k = i & 127;
            smem[OFF_PATH + i] += smem[OFF_WPREV + n] * smem[OFF_H + n * 130 + k];
        }
        __syncthreads();
        if (t == TS) break;   // final round only accumulates the last gather

        // ---- policy hidden: P1 = relu(H @ pol1_w^T + pol1_b)   (WMMA f32)
        gemm32(&smem[OFF_H], 130, &smem[OFF_POL1W], 130, HD,
               &smem[OFF_P1], 130, &smem[OFF_POL1B], 1, tid);
        __syncthreads();

        // ---- logits + gumbel: 32x9 outputs
        for (int i = tid; i < NS * 9; i += 256) {
            const int n = i / 9, j = i % 9;
            float a = smem[OFF_POL2B + j];
            const float* pr = &smem[OFF_P1 + n * 130];
            const float* wr = &smem[OFF_POL2W + j * HD];
            for (int k = 0; k < HD; ++k) a += pr[k] * wr[k];
            const float u = gu[(size_t)(t * NS + n) * 9 + j];
            const float g = -logf(-logf(u + 1e-10f) + 1e-10f);
            smem[OFF_LOGIT + n * 12 + j] = a + g;
        }
        __syncthreads();

        // ---- per-sample decision
        if (tid < NS) {
            float l[9];
            float mx = -1e30f;
            for (int j = 0; j < 9; ++j) { l[j] = smem[OFF_LOGIT + tid * 12 + j]; mx = fmaxf(mx, l[j]); }
            float den = 0.f, e8 = 0.f;
            for (int j = 0; j < 9; ++j) {
                const float e = expf(l[j] - mx);
                den += e;
                if (j == 8) e8 = e;
            }
            const float stop = e8 / den;
            int am = 0; float best = l[0];
            for (int j = 1; j < 8; ++j) if (l[j] > best) { best = l[j]; am = j; }
            const int pos = currL[tid];
            const int py = pos / 56, px = pos % 56;
            const int ny = min(max(py + dy[am], 0), 55);
            const int nx = min(max(px + dx[am], 0), 55);
            int nxt = ny * 56 + nx;
            if ((visL[tid * 98 + (nxt >> 5)] >> (nxt & 31)) & 1u) nxt = pos;
            if (stop != 0.0f) nxt = pos;
            visL[tid * 98 + (nxt >> 5)] |= (1u << (nxt & 31));
            currL[tid] = nxt;
            smem[OFF_WPREV + tid] = 1.f - stop;
        }
        __syncthreads();
    }

    // ---- att_dfs + fused attention factor
    const float fw0 = fusion_w[0], fw1 = fusion_w[1];
    const float fm = fmaxf(fw0, fw1);
    const float e0 = expf(fw0 - fm), e1 = expf(fw1 - fm);
    const float nw0 = e0 / (e0 + e1), nw1 = e1 / (e0 + e1);

    for (int i = tid; i < NS * CH; i += 256) {
        const int n = i >> 8, c = i & 255;
        const float* pr = &smem[OFF_PATH + n * HD];
        const float* wr = dfs_w + c * HD;
        float dot = 0.f;
        for (int k = 0; k < HD; ++k) dot += pr[k] * wr[k];
        const float attdfs = sigmoidf(dfs_b[c] + dot * (1.f / (float)NP));
        att_total_g[i] = nw0 * (0.5f * attse_g[i] + 0.5f * attdfs) + nw1 * attse2_g[i];
    }
}

// ---------------------------------------------------------------------------
// Kernel D: out = x * att_total.  x reads hit the 192MB L2 (resident from
// k_means); output uses non-temporal stores (native clang vector type) so
// the 103MB write stream does not evict x mid-kernel.
// ---------------------------------------------------------------------------
__global__ void k_scale(const float* __restrict__ x,
                        const float* __restrict__ att_total_g,
                        float* __restrict__ out) {
    const int row = blockIdx.x;                 // n*256 + c
    const float at = att_total_g[row];
    const v4f* xr = (const v4f*)(x + (size_t)row * NP);
    v4f* orw = (v4f*)(out + (size_t)row * NP);
    for (int i = threadIdx.x; i < NP / 4; i += 256) {
        v4f v = xr[i] * at;
        __builtin_nontemporal_store(v, &orw[i]);
    }
}

// ---------------------------------------------------------------------------
extern "C" void kernel_launch(void* const* d_in, const int* in_sizes, int n_in,
                              void* d_out, int out_size, void* d_ws, size_t ws_size,
                              hipStream_t stream) {
    const float* x      = (const float*)d_in[0];
    const float* gu     = (const float*)d_in[1];
    const float* se1_w  = (const float*)d_in[2];
    const float* se1_b  = (const float*)d_in[3];
    const float* se2_w  = (const float*)d_in[4];
    const float* se2_b  = (const float*)d_in[5];
    const float* node_w = (const float*)d_in[6];
    const float* node_b = (const float*)d_in[7];
    const float* pol1_w = (const float*)d_in[8];
    const float* pol1_b = (const float*)d_in[9];
    const float* pol2_w = (const float*)d_in[10];
    const float* pol2_b = (const float*)d_in[11];
    const float* dfs_w  = (const float*)d_in[12];
    const float* dfs_b  = (const float*)d_in[13];
    const float* bse1_w = (const float*)d_in[14];
    const float* bse1_b = (const float*)d_in[15];
    const float* bse2_w = (const float*)d_in[16];
    const float* bse2_b = (const float*)d_in[17];
    const float* fw     = (const float*)d_in[18];
    float* out = (float*)d_out;

    float* ws      = (float*)d_ws;
    float* s_g     = ws;               // 8192
    float* attse   = ws + 8192;        // 8192
    float* attse2  = ws + 16384;       // 8192
    float* att_tot = ws + 24576;       // 8192

    k_means<<<NS * CH, 256, 0, stream>>>(x, s_g);
    k_se<<<NS, 256, 0, stream>>>(s_g, se1_w, se1_b, se2_w, se2_b,
                                 bse1_w, bse1_b, bse2_w, bse2_b, attse, attse2);
    k_scan<<<1, 256, SMEM_BYTES, stream>>>(x, gu, node_w, node_b, pol1_w, pol1_b,
                                           pol2_w, pol2_b, dfs_w, dfs_b, fw,
                                           attse, attse2, att_tot);
    k_scale<<<NS * CH, 256, 0, stream>>>(x, att_tot, out);
}